// SingleAttentionHead_79413945303616
// MI455X (gfx1250) — compile-verified
//
#include <hip/hip_runtime.h>
#include <hip/hip_bf16.h>
#include <math.h>

#define B_  32
#define S_  2048
#define D_  1024
#define KQ_ 256
#define OUT_ 256
#define NCHUNK 16
#define CHUNK (S_ / NCHUNK)   // 128

typedef float v2f __attribute__((ext_vector_type(2)));
typedef float v8f __attribute__((ext_vector_type(8)));

// ---------------------------------------------------------------------------
// Small f32 GEMM via V_WMMA_F32_16X16X4_F32. One wave (32 threads) computes a
// 16x16 C tile. M is a multiple of 16 (here 32). BT=true means C = A * B^T
// with B stored [N,K] row-major (used for q @ Wk^T).
// Lane layouts per CDNA5 ISA 7.12.2:
//   A 16x4 : lane L: row M=L%16 ; VGPR0=K0, VGPR1=K0+1 with K0 = (L<16)?0:2
//   B 4x16 : lane L: col N=L%16 ; VGPR0=K0, VGPR1=K0+1 (same K split)
//   C 16x16: VGPR r : lanes0-15 -> M=r, lanes16-31 -> M=r+8 ; N = L%16
// ---------------------------------------------------------------------------
template <bool BT>
__global__ void gemm_f32_wmma_kernel(const float* __restrict__ A,
                                     const float* __restrict__ B,
                                     float* __restrict__ C,
                                     int N, int K, int lda, int ldb, int ldc) {
  const int ntn  = N >> 4;
  const int tile = blockIdx.x;
  const int tm   = (tile / ntn) << 4;
  const int tn   = (tile % ntn) << 4;
  const int lane = threadIdx.x & 31;
  const int half = lane >> 4;        // 0: K lanes {0,1}, 1: K lanes {2,3}
  const int l16  = lane & 15;

  v8f acc = {};
  for (int kk = 0; kk < K; kk += 4) {
    const int k0 = kk + half * 2;
    v2f a, b;
    a.x = A[(size_t)(tm + l16) * lda + k0];
    a.y = A[(size_t)(tm + l16) * lda + k0 + 1];
    if (BT) {
      b.x = B[(size_t)(tn + l16) * ldb + k0];
      b.y = B[(size_t)(tn + l16) * ldb + k0 + 1];
    } else {
      b.x = B[(size_t)(k0)     * ldb + tn + l16];
      b.y = B[(size_t)(k0 + 1) * ldb + tn + l16];
    }
    acc = __builtin_amdgcn_wmma_f32_16x16x4_f32(
        /*neg_a=*/false, a, /*neg_b=*/false, b,
        /*c_mod=*/(short)0, acc, /*reuse_a=*/false, /*reuse_b=*/false);
  }
#pragma unroll
  for (int r = 0; r < 8; ++r)
    C[(size_t)(tm + half * 8 + r) * ldc + tn + l16] = acc[r];
}

// ---------------------------------------------------------------------------
// attn_raw[b,s] = scale * dot(keys[b,s,:], qk[b,:])   (D=1024)
// One wave per s-row; 32 lanes x float4 x 8 iters = 1024 elements.
// grid: (S/8, B), block: 256 (8 waves).
// ---------------------------------------------------------------------------
__global__ void scores_kernel(const float* __restrict__ keys,
                              const float* __restrict__ qk,
                              float* __restrict__ raw) {
  const int b    = blockIdx.y;
  const int wv   = threadIdx.x >> 5;
  const int lane = threadIdx.x & 31;
  const int s    = blockIdx.x * 8 + wv;

  const float4* krow = (const float4*)(keys + ((size_t)b * S_ + s) * D_);
  const float4* qrow = (const float4*)(qk + (size_t)b * D_);
  float acc = 0.f;
#pragma unroll
  for (int i = 0; i < 8; ++i) {
    float4 kv = krow[i * 32 + lane];
    float4 qv = qrow[i * 32 + lane];
    acc += kv.x * qv.x + kv.y * qv.y + kv.z * qv.z + kv.w * qv.w;
  }
#pragma unroll
  for (int o = 16; o > 0; o >>= 1) acc += __shfl_down(acc, o, 32);
  if (lane == 0) raw[(size_t)b * S_ + s] = acc * 0.0625f;  // 1/sqrt(256)
}

// ---------------------------------------------------------------------------
// Row softmax over S=2048 (max-subtract, matches jax.nn.softmax).
// grid: B blocks of 256 threads.
// ---------------------------------------------------------------------------
__global__ void softmax_kernel(const float* __restrict__ raw,
                               float* __restrict__ w) {
  __shared__ float red[8];
  __shared__ float bval;
  const int b    = blockIdx.x;
  const int tid  = threadIdx.x;
  const int lane = tid & 31, wv = tid >> 5;
  const float* row = raw + (size_t)b * S_;

  float m = -3.402823466e38f;
  for (int s = tid; s < S_; s += 256) m = fmaxf(m, row[s]);
#pragma unroll
  for (int o = 16; o > 0; o >>= 1) m = fmaxf(m, __shfl_xor(m, o, 32));
  if (lane == 0) red[wv] = m;
  __syncthreads();
  if (tid == 0) {
    float mm = red[0];
#pragma unroll
    for (int i = 1; i < 8; ++i) mm = fmaxf(mm, red[i]);
    bval = mm;
  }
  __syncthreads();
  const float gm = bval;

  float sum = 0.f;
  for (int s = tid; s < S_; s += 256) sum += expf(row[s] - gm);
#pragma unroll
  for (int o = 16; o > 0; o >>= 1) sum += __shfl_xor(sum, o, 32);
  __syncthreads();
  if (lane == 0) red[wv] = sum;
  __syncthreads();
  if (tid == 0) {
    float ss = 0.f;
#pragma unroll
    for (int i = 0; i < 8; ++i) ss += red[i];
    bval = ss;
  }
  __syncthreads();
  const float inv = 1.0f / bval;
  for (int s = tid; s < S_; s += 256)
    w[(size_t)b * S_ + s] = expf(row[s] - gm) * inv;
}

// ---------------------------------------------------------------------------
// part[c, b, d] = sum over s-chunk c of w[b,s] * values[b,s,d]
// Each block: one (b, chunk). 256 threads x float4 covers d = 0..1023 exactly.
// grid: (NCHUNK, B). Deterministic (no atomics).
// ---------------------------------------------------------------------------
__global__ void ctxd_partial_kernel(const float* __restrict__ values,
                                    const float* __restrict__ w,
                                    float* __restrict__ part) {
  const int b = blockIdx.y, c = blockIdx.x, tid = threadIdx.x;
  const float* wrow = w + (size_t)b * S_ + c * CHUNK;
  float4 acc = make_float4(0.f, 0.f, 0.f, 0.f);
  for (int i = 0; i < CHUNK; ++i) {
    const float wt = wrow[i];
    const float4* vrow =
        (const float4*)(values + ((size_t)b * S_ + (size_t)c * CHUNK + i) * D_);
    float4 v = vrow[tid];
    acc.x += wt * v.x; acc.y += wt * v.y;
    acc.z += wt * v.z; acc.w += wt * v.w;
  }
  float4* dst = (float4*)(part + (size_t)c * (B_ * D_) + (size_t)b * D_) + tid;
  *dst = acc;
}

// ctxd[b*D+d] = sum over chunks of part[c][b*D+d]   (32768 elements)
__global__ void reduce_partials_kernel(const float* __restrict__ part,
                                       float* __restrict__ ctxd) {
  const int idx = blockIdx.x * 256 + threadIdx.x;  // 0..32767
  float s = 0.f;
#pragma unroll
  for (int c = 0; c < NCHUNK; ++c) s += part[(size_t)c * (B_ * D_) + idx];
  ctxd[idx] = s;
}

extern "C" void kernel_launch(void* const* d_in, const int* in_sizes, int n_in,
                              void* d_out, int out_size, void* d_ws,
                              size_t ws_size, hipStream_t stream) {
  const float* queries = (const float*)d_in[0];  // [32,1024]
  const float* keys    = (const float*)d_in[1];  // [32,2048,1024]
  const float* values  = (const float*)d_in[2];  // [32,2048,1024]
  const float* Wq      = (const float*)d_in[3];  // [1024,256]
  const float* Wk      = (const float*)d_in[4];  // [1024,256]
  const float* Wv      = (const float*)d_in[5];  // [1024,256]

  float* out     = (float*)d_out;
  float* context = out;                 // [32,1,256]  -> 8192 floats
  float* attn_w  = out + B_ * OUT_;     // [32,1,2048] -> 65536 floats

  float* ws   = (float*)d_ws;
  float* raw  = ws;                       // 65536 floats: raw scores
  float* q    = raw + B_ * S_;            // 8192:  q = queries @ Wq
  float* qk   = q + B_ * KQ_;             // 32768: qk = q @ Wk^T
  float* ctxd = qk + B_ * D_;             // 32768: weighted value row-sum
  float* part = ctxd + B_ * D_;           // 524288: per-chunk partials

  // 1) q = queries @ Wq          [32,1024]x[1024,256] -> [32,256]
  gemm_f32_wmma_kernel<false><<<(B_ / 16) * (KQ_ / 16), 32, 0, stream>>>(
      queries, Wq, q, KQ_, D_, D_, KQ_, KQ_);

  // 2) qk = q @ Wk^T             [32,256]x[256,1024] -> [32,1024]
  gemm_f32_wmma_kernel<true><<<(B_ / 16) * (D_ / 16), 32, 0, stream>>>(
      q, Wk, qk, D_, KQ_, KQ_, KQ_, D_);

  // 3) raw[b,s] = scale * keys[b,s,:] . qk[b,:]
  scores_kernel<<<dim3(S_ / 8, B_), 256, 0, stream>>>(keys, qk, raw);

  // 4) attn_w = softmax(raw) written straight to output
  softmax_kernel<<<B_, 256, 0, stream>>>(raw, attn_w);

  // 5) per-chunk weighted row sums of values, then deterministic reduce
  ctxd_partial_kernel<<<dim3(NCHUNK, B_), 256, 0, stream>>>(values, attn_w,
                                                            part);
  reduce_partials_kernel<<<(B_ * D_) / 256, 256, 0, stream>>>(part, ctxd);

  // 6) context = ctxd @ Wv       [32,1024]x[1024,256] -> [32,256]
  gemm_f32_wmma_kernel<false><<<(B_ / 16) * (OUT_ / 16), 32, 0, stream>>>(
      ctxd, Wv, context, OUT_, D_, D_, OUT_, OUT_);
}